// BEVWarp_6828998000690
// MI455X (gfx1250) — compile-verified
//
#include <hip/hip_runtime.h>
#include <hip/hip_bf16.h>

// ---- problem constants (from the reference) ----
#define BB    2
#define VV    6
#define CLCH  256
#define HB    180
#define WB    180
#define IH    112
#define IW    200
#define NPTS  32768
#define ORI_H 448
#define ORI_W 800
#define EPSV  1e-5f

#define PIX_PER_VIEW (IH * IW)          // 22400
#define NVIEWIMG     (BB * VV)          // 12
#define DMELEMS      (NVIEWIMG * PIX_PER_VIEW) // 268800

typedef __attribute__((ext_vector_type(2))) float v2f;
typedef __attribute__((ext_vector_type(8))) float v8f;

// ------------------------------------------------------------------
// 0) zero the depth maps (ws is poisoned, must re-init every call)
// ------------------------------------------------------------------
__global__ void zero_kernel(float* __restrict__ p, int n) {
    int i = blockIdx.x * blockDim.x + threadIdx.x;
    if (i < n) p[i] = 0.0f;
}

// ------------------------------------------------------------------
// 1) invert the 12 lidar2img 4x4 matrices (Gauss-Jordan w/ pivoting)
// ------------------------------------------------------------------
__global__ void invert4x4_kernel(const float* __restrict__ l2i,
                                 float* __restrict__ i2l, int nmats) {
    int m = blockIdx.x * blockDim.x + threadIdx.x;
    if (m >= nmats) return;
    float a[4][8];
    for (int i = 0; i < 4; ++i)
        for (int j = 0; j < 4; ++j) {
            a[i][j]     = l2i[m * 16 + i * 4 + j];
            a[i][j + 4] = (i == j) ? 1.0f : 0.0f;
        }
    for (int col = 0; col < 4; ++col) {
        int piv = col; float best = fabsf(a[col][col]);
        for (int r = col + 1; r < 4; ++r) {
            float v = fabsf(a[r][col]);
            if (v > best) { best = v; piv = r; }
        }
        if (piv != col)
            for (int j = 0; j < 8; ++j) {
                float t = a[col][j]; a[col][j] = a[piv][j]; a[piv][j] = t;
            }
        float inv = 1.0f / a[col][col];
        for (int j = 0; j < 8; ++j) a[col][j] *= inv;
        for (int r = 0; r < 4; ++r) {
            if (r == col) continue;
            float f = a[r][col];
            for (int j = 0; j < 8; ++j) a[r][j] -= f * a[col][j];
        }
    }
    for (int i = 0; i < 4; ++i)
        for (int j = 0; j < 4; ++j)
            i2l[m * 16 + i * 4 + j] = a[i][j + 4];
}

// ------------------------------------------------------------------
// 2) project lidar points into each view, scatter depth (last-write
//    nondeterminism matches the reference's unspecified scatter order)
// ------------------------------------------------------------------
__global__ void scatter_depth_kernel(const float* __restrict__ pts,
                                     const float* __restrict__ l2i,
                                     float* __restrict__ dm) {
    int idx = blockIdx.x * blockDim.x + threadIdx.x;
    if (idx >= BB * NPTS) return;
    int b = idx / NPTS, n = idx % NPTS;
    float px = pts[(size_t)(b * NPTS + n) * 3 + 0];
    float py = pts[(size_t)(b * NPTS + n) * 3 + 1];
    float pz = pts[(size_t)(b * NPTS + n) * 3 + 2];
    #pragma unroll
    for (int v = 0; v < VV; ++v) {
        const float* M = l2i + (size_t)(b * VV + v) * 16;
        float X  = M[0] * px + M[1] * py + M[2]  * pz + M[3];
        float Y  = M[4] * px + M[5] * py + M[6]  * pz + M[7];
        float Zc = M[8] * px + M[9] * py + M[10] * pz + M[11];
        if (Zc > EPSV) {
            float z = fmaxf(Zc, EPSV);
            float x = X / z, y = Y / z;
            // px,py in (-1,1)  <=>  0 < x < ORI_W, 0 < y < ORI_H
            if (x > 0.0f && x < (float)ORI_W && y > 0.0f && y < (float)ORI_H) {
                int xi = (int)(x * ((float)IW / (float)ORI_W));
                int yi = (int)(y * ((float)IH / (float)ORI_H));
                if (xi >= 0 && xi < IW && yi >= 0 && yi < IH)
                    dm[((size_t)(b * VV + v) * IH + yi) * IW + xi] = Zc;
            }
        }
    }
}

// ------------------------------------------------------------------
// 3) one masked max-dilation pass (reference _fill_depth step),
//    LDS-tiled: 16x16 block, halo up to 4 -> 24x24 tile
// ------------------------------------------------------------------
__global__ void fill_depth_kernel(const float* __restrict__ in,
                                  float* __restrict__ out, int radius) {
    __shared__ float tile[24 * 24];
    const int bv  = blockIdx.z;
    const int tw  = 16 + 2 * radius;
    const int bx0 = blockIdx.x * 16 - radius;
    const int by0 = blockIdx.y * 16 - radius;
    const float* img = in + (size_t)bv * PIX_PER_VIEW;
    int tid = threadIdx.y * 16 + threadIdx.x;
    for (int t = tid; t < tw * tw; t += 256) {
        int ty = t / tw, tx = t % tw;
        int gy = by0 + ty, gx = bx0 + tx;
        float val = 0.0f;
        if (gx >= 0 && gx < IW && gy >= 0 && gy < IH) val = img[gy * IW + gx];
        tile[t] = val;
    }
    __syncthreads();
    int w = blockIdx.x * 16 + threadIdx.x;
    int h = blockIdx.y * 16 + threadIdx.y;
    if (w >= IW || h >= IH) return;
    float center = tile[(threadIdx.y + radius) * tw + (threadIdx.x + radius)];
    float res = center;
    if (center <= 0.0f) {                       // fill only empty pixels
        float mx = 0.0f;
        for (int dy = -radius; dy <= radius; ++dy)
            for (int dx = -radius; dx <= radius; ++dx)
                mx = fmaxf(mx, tile[(threadIdx.y + radius + dy) * tw +
                                    (threadIdx.x + radius + dx)]);
        res = mx;
    }
    out[(size_t)bv * PIX_PER_VIEW + h * IW + w] = res;
}

// ------------------------------------------------------------------
// 4) lift depth-map pixels to lidar frame with V_WMMA_F32_16X16X4_F32.
//    One wave lifts 16 pixels per WMMA: D(16x16) = A(16x4) * B(4x16)
//      A: i2l rows replicated (row = lane&3). ISA A layout (16x4 f32):
//         lanes 0-15 VGPR{0,1} = K{0,1}; lanes 16-31 VGPR{0,1} = K{2,3}
//      B: 16 pixel homogeneous coords (x*d, y*d, d, 1) as columns;
//         mirrored layout: lanes 0-15 hold (K0,K1), lanes 16-31 (K2,K3)
//      D rows 0..2 (VGPR 0..2, lanes 0-15) = lifted xyz of pixel j=lane.
//    Branch-free operand setup keeps EXEC all-1s at the WMMA.
// ------------------------------------------------------------------
__global__ void lift_wmma_kernel(const float* __restrict__ dm,
                                 const float* __restrict__ i2l,
                                 float4* __restrict__ uvm) {
    const int lane  = threadIdx.x & 31;
    const int wave  = threadIdx.x >> 5;
    const int group = blockIdx.x * (blockDim.x >> 5) + wave;  // 16 px each
    const int groupsPerView = PIX_PER_VIEW / 16;              // 1400
    if (group >= NVIEWIMG * groupsPerView) return;            // wave-uniform
    const int bv  = group / groupsPerView;
    const int pix = (group % groupsPerView) * 16 + (lane & 15);
    const int h = pix / IW, w = pix % IW;

    const float d  = dm[(size_t)bv * PIX_PER_VIEW + pix];
    const float xg = (float)w * ((float)(ORI_W - 1) / (float)(IW - 1));
    const float yg = (float)h * ((float)(ORI_H - 1) / (float)(IH - 1));
    const bool lo = lane < 16;

    v2f bmat;
    bmat.x = lo ? xg * d : d;
    bmat.y = lo ? yg * d : 1.0f;

    const float* M = i2l + (size_t)bv * 16;
    const int r = lane & 3;
    v2f amat;
    amat.x = lo ? M[r * 4 + 0] : M[r * 4 + 2];
    amat.y = lo ? M[r * 4 + 1] : M[r * 4 + 3];

    v8f cacc = {};
    v8f dres = __builtin_amdgcn_wmma_f32_16x16x4_f32(
        false, amat, false, bmat, (short)0, cacc, false, false);

    if (lo) {
        float X = dres[0], Y = dres[1], Z = dres[2];
        bool inr = (X > -54.0f) && (X < 54.0f) &&
                   (Y > -54.0f) && (Y < 54.0f) &&
                   (Z > -5.0f)  && (Z < 3.0f);
        float u = ((X + 54.0f) * (1.0f / 108.0f) - 0.5f) * 2.0f;
        float v = ((Y + 54.0f) * (1.0f / 108.0f) - 0.5f) * 2.0f;
        uvm[(size_t)bv * PIX_PER_VIEW + pix] =
            make_float4(u, v, inr ? 1.0f : 0.0f, 0.0f);
    }
}

// ------------------------------------------------------------------
// 5) bilinear sample 256 BEV channels per pixel.
//    - corner offsets + (zero-pad x lift) masks folded into 4 weights
//      computed once, reused for all 256 channels
//    - output stores are NON-TEMPORAL: the 275 MB write-once stream
//      must not evict the 66 MB L2-resident BEV feature map
//    - uvm packets are read-once -> non-temporal loads
//    - light global_prefetch of the next channel planes' corner lines
// ------------------------------------------------------------------
__global__ void sample_kernel(const float* __restrict__ lidar_feats,
                              const float4* __restrict__ uvm,
                              float* __restrict__ out) {
    int idx = blockIdx.x * blockDim.x + threadIdx.x;
    if (idx >= DMELEMS) return;
    int bv  = idx / PIX_PER_VIEW;
    int pix = idx % PIX_PER_VIEW;
    int b   = bv / VV;

    const float* up = (const float*)(uvm + idx);
    float gx = __builtin_nontemporal_load(up + 0);
    float gy = __builtin_nontemporal_load(up + 1);
    float gm = __builtin_nontemporal_load(up + 2);

    float fx = (gx + 1.0f) * 0.5f * (float)WB - 0.5f;
    float fy = (gy + 1.0f) * 0.5f * (float)HB - 0.5f;
    float x0f = floorf(fx), y0f = floorf(fy);
    float wx1 = fx - x0f, wy1 = fy - y0f;
    float wx0 = 1.0f - wx1, wy0 = 1.0f - wy1;

    float vx0 = (x0f >= 0.0f        && x0f <= (float)(WB - 1)) ? 1.0f : 0.0f;
    float vx1 = (x0f + 1.0f >= 0.0f && x0f + 1.0f <= (float)(WB - 1)) ? 1.0f : 0.0f;
    float vy0 = (y0f >= 0.0f        && y0f <= (float)(HB - 1)) ? 1.0f : 0.0f;
    float vy1 = (y0f + 1.0f >= 0.0f && y0f + 1.0f <= (float)(HB - 1)) ? 1.0f : 0.0f;

    int x0 = min(max((int)x0f, 0), WB - 1);
    int x1 = min(max((int)x0f + 1, 0), WB - 1);
    int y0 = min(max((int)y0f, 0), HB - 1);
    int y1 = min(max((int)y0f + 1, 0), HB - 1);

    float w00 = wx0 * wy0 * vx0 * vy0 * gm;
    float w01 = wx1 * wy0 * vx1 * vy0 * gm;
    float w10 = wx0 * wy1 * vx0 * vy1 * gm;
    float w11 = wx1 * wy1 * vx0 * vy1 * gm;  // corrected below
    w11 = wx1 * wy1 * vx1 * vy1 * gm;

    int o00 = y0 * WB + x0, o01 = y0 * WB + x1;
    int o10 = y1 * WB + x0, o11 = y1 * WB + x1;

    const float* L = lidar_feats + (size_t)b * CLCH * HB * WB;
    float* o = out + (size_t)bv * CLCH * PIX_PER_VIEW + pix;
    #pragma unroll 4
    for (int c = 0; c < CLCH; ++c) {
        const float* Lc = L + (size_t)c * (HB * WB);
        if ((c & 15) == 0 && c + 16 < CLCH) {
            // warm the corner cachelines of the plane 16 channels ahead
            const float* Lp = L + (size_t)(c + 16) * (HB * WB);
            __builtin_prefetch(Lp + o00, 0, 3);
            __builtin_prefetch(Lp + o10, 0, 3);
        }
        float acc = w00 * Lc[o00] + w01 * Lc[o01] +
                    w10 * Lc[o10] + w11 * Lc[o11];
        __builtin_nontemporal_store(acc, o + (size_t)c * PIX_PER_VIEW);
    }
}

// ------------------------------------------------------------------
extern "C" void kernel_launch(void* const* d_in, const int* in_sizes, int n_in,
                              void* d_out, int out_size, void* d_ws, size_t ws_size,
                              hipStream_t stream) {
    const float* lidar_feats = (const float*)d_in[0];   // [B,CL,HB,WB]
    // d_in[1] = img_feats: shape-only in the reference, unused
    const float* pts = (const float*)d_in[2];           // [B,NPTS,3]
    const float* l2i = (const float*)d_in[3];           // [B,V,4,4]
    float* out = (float*)d_out;                         // [B,V,CL,IH,IW]

    // workspace layout
    float*  dmA = (float*)d_ws;                 // 268800 f32
    float*  dmB = dmA + DMELEMS;                // 268800 f32
    float*  i2l = dmB + DMELEMS;                // 192 f32
    float4* uvm = (float4*)(i2l + NVIEWIMG * 16);  // 268800 float4 (16B aligned)

    // 0) init depth maps (ws is not re-poisoned between replays)
    zero_kernel<<<(DMELEMS + 255) / 256, 256, 0, stream>>>(dmA, DMELEMS);

    // 1) invert the 12 camera matrices
    invert4x4_kernel<<<1, 16, 0, stream>>>(l2i, i2l, NVIEWIMG);

    // 2) scatter point depths into per-view depth maps
    scatter_depth_kernel<<<(BB * NPTS + 255) / 256, 256, 0, stream>>>(pts, l2i, dmA);

    // 3) four masked dilation passes k = 3,5,7,9 (ping-pong A<->B)
    dim3 fgrid((IW + 15) / 16, (IH + 15) / 16, NVIEWIMG);
    dim3 fblk(16, 16, 1);
    fill_depth_kernel<<<fgrid, fblk, 0, stream>>>(dmA, dmB, 1);
    fill_depth_kernel<<<fgrid, fblk, 0, stream>>>(dmB, dmA, 2);
    fill_depth_kernel<<<fgrid, fblk, 0, stream>>>(dmA, dmB, 3);
    fill_depth_kernel<<<fgrid, fblk, 0, stream>>>(dmB, dmA, 4);

    // 4) WMMA lift: 16 pixels per wave per v_wmma_f32_16x16x4_f32
    const int totalGroups = NVIEWIMG * (PIX_PER_VIEW / 16);   // 16800
    const int wavesPerBlk = 4;                                // 128 threads
    lift_wmma_kernel<<<totalGroups / wavesPerBlk, wavesPerBlk * 32, 0, stream>>>(
        dmA, i2l, uvm);

    // 5) bilinear sample all 256 channels, NT stores + prefetch
    sample_kernel<<<(DMELEMS + 255) / 256, 256, 0, stream>>>(lidar_feats, uvm, out);
}